// GIN_23270132810411
// MI455X (gfx1250) — compile-verified
//
#include <hip/hip_runtime.h>
#include <hip/hip_bf16.h>

typedef float v2f __attribute__((ext_vector_type(2)));
typedef float v8f __attribute__((ext_vector_type(8)));

// ---------------- zero fill (agg buffers must be zeroed every call) -------
__global__ void gin_fill_zero(float* __restrict__ p, long n) {
  long i = (long)blockIdx.x * blockDim.x + threadIdx.x;
  long stride = (long)gridDim.x * blockDim.x;
  for (; i < n; i += stride) p[i] = 0.0f;
}

// ---------------- edge scatter-add: agg[dst] += h[src] --------------------
// one thread = (edge, 4-feature chunk); float4 gather + 4 f32 atomics
__global__ void gin_scatter_add(const float* __restrict__ h,
                                const long long* __restrict__ src,
                                const long long* __restrict__ dst,
                                float* __restrict__ agg, int E) {
  long t = (long)blockIdx.x * blockDim.x + threadIdx.x;
  long total = (long)E * 16;
  if (t >= total) return;
  int e = (int)(t >> 4);
  int c = (int)(t & 15);
  int s = (int)src[e];
  int d = (int)dst[e];
  const float4 v = *(const float4*)(h + (long)s * 64 + c * 4);
  float* ap = agg + (long)d * 64 + c * 4;
  unsafeAtomicAdd(ap + 0, v.x);
  unsafeAtomicAdd(ap + 1, v.y);
  unsafeAtomicAdd(ap + 2, v.z);
  unsafeAtomicAdd(ap + 3, v.w);
}

// ---------------- fused (X [+ AGG]) @ W + b, optional ReLU ----------------
// K = 64 fixed, W is [64 x nout] row-major. Each wave owns a 16-row M tile
// and NTILES 16-col N tiles. W staged transposed in LDS: WT[col*68 + k]
// (stride 68 -> ds_load_b64 per WMMA operand, 64-bank conflict-free).
// NTILES is a template param so all acc[] indexing is static (no v_movrel).
template <int NTILES, bool HAS_AGG, bool RELU>
__global__ __launch_bounds__(256) void gin_gemm(
    const float* __restrict__ X, const float* __restrict__ AGG,
    const float* __restrict__ W, const float* __restrict__ B,
    float* __restrict__ OUT, int M, int nout) {
  constexpr int LDN = NTILES * 16;   // padded col count
  constexpr int KS  = 68;            // LDS k-stride (pad for bank spread)
  __shared__ float WT[64 * KS];      // [col][k], col-major-transposed W
  __shared__ float Blds[64];

  for (int i = threadIdx.x; i < 64 * LDN; i += 256) {
    int k = i / LDN, n = i - k * LDN;
    WT[n * KS + k] = (n < nout) ? W[k * nout + n] : 0.0f;
  }
  for (int i = threadIdx.x; i < LDN; i += 256)
    Blds[i] = (i < nout) ? B[i] : 0.0f;
  __syncthreads();

  const int lane = threadIdx.x & 31;
  const int half = lane >> 4;        // 0: K {0,1} of each K4 step, 1: K {2,3}
  const int l15  = lane & 15;
  const int m0 = (blockIdx.x * 8 + (threadIdx.x >> 5)) * 16;
  if (m0 >= M) return;               // wave-uniform: EXEC stays all-ones

  // A tile 16x64 in registers, WMMA f32 16x16x4 layout
  int row = m0 + l15;
  int rc = row < M ? row : (M - 1);  // clamp, keeps all lanes active
  const float* xr = X + (long)rc * 64;
  const float* ar = HAS_AGG ? AGG + (long)rc * 64 : nullptr;
  v2f a[16];
#pragma unroll
  for (int s = 0; s < 16; ++s) {
    int k0 = 4 * s + 2 * half;
    float a0 = xr[k0], a1 = xr[k0 + 1];
    if (HAS_AGG) { a0 += ar[k0]; a1 += ar[k0 + 1]; }
    a[s][0] = a0; a[s][1] = a1;
  }

  v8f acc[NTILES];
#pragma unroll
  for (int t = 0; t < NTILES; ++t) acc[t] = {};

#pragma unroll
  for (int s = 0; s < 16; ++s) {
    const int k0 = 4 * s + 2 * half;
#pragma unroll
    for (int t = 0; t < NTILES; ++t) {
      const int col = t * 16 + l15;
      v2f bm = *(const v2f*)&WT[col * KS + k0];   // ds_load_b64
      acc[t] = __builtin_amdgcn_wmma_f32_16x16x4_f32(
          false, a[s], false, bm, (short)0, acc[t], false, false);
    }
  }

#pragma unroll
  for (int t = 0; t < NTILES; ++t) {
    const int col = t * 16 + l15;
    float bias = Blds[col];
#pragma unroll
    for (int v = 0; v < 8; ++v) {
      int r = m0 + v + 8 * half;
      if (r < M && col < nout) {
        float val = acc[t][v] + bias;
        if (RELU) val = fmaxf(val, 0.0f);
        OUT[(long)r * nout + col] = val;
      }
    }
  }
}

// ---------------- row-wise log_softmax over 40 classes --------------------
__global__ void gin_log_softmax40(const float* __restrict__ logits,
                                  float* __restrict__ out, int M) {
  int r = blockIdx.x * blockDim.x + threadIdx.x;
  if (r >= M) return;
  const float* p = logits + (long)r * 40;
  float m = -3.4e38f;
  for (int j = 0; j < 40; ++j) m = fmaxf(m, p[j]);
  float s = 0.0f;
  for (int j = 0; j < 40; ++j) s += __expf(p[j] - m);
  float lse = m + __logf(s);
  float* o = out + (long)r * 40;
  for (int j = 0; j < 40; ++j) o[j] = p[j] - lse;
}

extern "C" void kernel_launch(void* const* d_in, const int* in_sizes, int n_in,
                              void* d_out, int out_size, void* d_ws, size_t ws_size,
                              hipStream_t stream) {
  const float*     x   = (const float*)d_in[0];
  const long long* ei  = (const long long*)d_in[1];
  const float*     W1  = (const float*)d_in[2];
  const float*     b1  = (const float*)d_in[3];
  const float*     W2  = (const float*)d_in[4];
  const float*     b2  = (const float*)d_in[5];
  const float*     Wf  = (const float*)d_in[6];
  const float*     bf  = (const float*)d_in[7];

  const int N = in_sizes[0] / 64;   // 100000 nodes
  const int E = in_sizes[1] / 2;    // 1600000 edges
  const long long* src = ei;
  const long long* dst = ei + E;

  char* ws = (char*)d_ws;
  const size_t slot = (size_t)N * 64 * sizeof(float);   // 25.6 MB
  float* agg    = (float*)ws;
  float* h1     = (float*)(ws + slot);
  float* h2     = (float*)(ws + 2 * slot);
  float* logits = agg;              // agg slot free after layer-2 GEMM

  dim3 blk(256);
  const long aggN = (long)N * 64;
  const int fillB = (int)((aggN + 255) / 256);
  const int scatB = (int)(((long)E * 16 + 255) / 256);
  const int gemmB = (N + 127) / 128;   // 8 waves/block * 16 rows/wave
  const int lsB   = (N + 255) / 256;

  // ---- layer 1: h1 = relu((x + segsum(x[src]->dst)) @ W1 + b1)
  gin_fill_zero<<<fillB, blk, 0, stream>>>(agg, aggN);
  gin_scatter_add<<<scatB, blk, 0, stream>>>(x, src, dst, agg, E);
  gin_gemm<4, true, true><<<gemmB, blk, 0, stream>>>(x, agg, W1, b1, h1, N, 64);

  // ---- layer 2: h2 = relu((h1 + segsum(h1[src]->dst)) @ W2 + b2)
  gin_fill_zero<<<fillB, blk, 0, stream>>>(agg, aggN);
  gin_scatter_add<<<scatB, blk, 0, stream>>>(h1, src, dst, agg, E);
  gin_gemm<4, true, true><<<gemmB, blk, 0, stream>>>(h1, agg, W2, b2, h2, N, 64);

  // ---- classifier + log_softmax
  gin_gemm<3, false, false><<<gemmB, blk, 0, stream>>>(h2, nullptr, Wf, bf, logits, N, 40);
  gin_log_softmax40<<<lsB, blk, 0, stream>>>(logits, (float*)d_out, N);
}